// snn_layer_conv_16947940950660
// MI455X (gfx1250) — compile-verified
//
#include <hip/hip_runtime.h>
#include <math.h>

typedef __attribute__((ext_vector_type(2))) float v2f;
typedef __attribute__((ext_vector_type(4))) float v4f;
typedef __attribute__((ext_vector_type(8))) float v8f;
typedef unsigned int u32x4 __attribute__((ext_vector_type(4)));
typedef int i32x4 __attribute__((ext_vector_type(4)));
typedef int i32x8 __attribute__((ext_vector_type(8)));

#define IH 128
#define IW 128
#define CI 32
#define CO 32
#define HO 126
#define WO 126
#define KSTEPS 72            // 288 / 4
#define TILE_COLS 66         // 64 outputs + 2 halo
#define TILE_STRIDE 36       // padded dword stride per (kh,col): bank-conflict-free + 16B aligned

#if __has_builtin(__builtin_amdgcn_tensor_load_to_lds)
#define HAVE_TDM 1
#else
#define HAVE_TDM 0
#endif

#if HAVE_TDM
__device__ __forceinline__ void tdm_load(u32x4 g0, i32x8 g1, i32x4 g2, i32x4 g3) {
#if __clang_major__ >= 23
  i32x8 z = {0, 0, 0, 0, 0, 0, 0, 0};
  __builtin_amdgcn_tensor_load_to_lds(g0, g1, g2, g3, z, 0);
#else
  __builtin_amdgcn_tensor_load_to_lds(g0, g1, g2, g3, 0);
#endif
}
#endif

// ---- weight_ring in the reference's exact f32 op order ----
__device__ __forceinline__ float weight_ring_f(float V) {
  const float pi   = 3.14159265358979323846f;  // rounds to f32 0x40490FDB like jnp.float32(math.pi)
  const float neff = 3.6f;
  const float wl   = 1.550183e-06f;
  const float Dd   = 0.0002f;
  const float Ls   = 0.000335841f;
  float Circ = pi * Dd + 2.0f * Ls;
  const float r = 0.1f, a = 0.1f;
  const float VLpi = 0.009f, Lp = 0.0002f;
  float Vpi   = rintf(VLpi / Lp);              // jnp.round -> 45
  float Vring = V * Vpi;
  float dneff = wl * Vring / (2.0f * Vpi);
  float dphi  = 2.0f * pi * dneff / wl;
  float phi   = 2.0f * pi * neff * Circ / wl + dphi;
  float c     = cosf(phi);
  float ar    = a * r;
  float num = a * a + r * r - 2.0f * a * r * c;
  float den = 1.0f + ar * ar - 2.0f * a * r * c;
  return num / den;
}

// ---- kernel 1: compute weight_ring(k) and pack into WMMA B-fragment layout ----
// frag index = kstep*64 + ntile*32 + lane ; each entry = float2 {W[K0+kk][N], W[K0+kk+1][N]}
// with kk = (lane/16)*2, N = ntile*16 + lane%16, K0 = kstep*4  (ISA f32 4-K operand layout)
__global__ void wring_pack_kernel(const float* __restrict__ k, v2f* __restrict__ bfrag) {
  int idx = blockIdx.x * blockDim.x + threadIdx.x;
  if (idx >= KSTEPS * 2 * 32) return;
  int lane  = idx & 31;
  int ntile = (idx >> 5) & 1;
  int kstep = idx >> 6;
  int K0 = kstep * 4 + ((lane >> 4) << 1);
  int N  = ntile * 16 + (lane & 15);
  v2f o;
  #pragma unroll
  for (int j = 0; j < 2; ++j) {
    int K  = K0 + j;
    int kh = K / 96;
    int rr = K % 96;
    int kw = rr >> 5;
    int ci = rr & 31;
    float V = k[(((kh * 3 + kw) * CI + ci) << 5) + N];
    o[j] = weight_ring_f(V);
  }
  bfrag[idx] = o;
}

// ---- kernel 2: implicit-GEMM conv via v_wmma_f32_16x16x4_f32 + fused 200-step SNN ----
__global__ void __launch_bounds__(128)
conv_snn_kernel(const float* __restrict__ inp, const v2f* __restrict__ bfrag,
                float* __restrict__ out) {
  __shared__ __align__(16) v2f   sB[KSTEPS * 2 * 32];                 // 36,864 B
  __shared__ __align__(16) float sX[3 * TILE_COLS * TILE_STRIDE];     // 28,512 B

  int bx    = blockIdx.x;
  int wquad = bx & 1;                 // which 64-wide output strip
  int t     = bx >> 1;
  int ho    = t % HO;
  int n     = t / HO;
  int c0    = wquad * 64;             // first output wo of this block
  int tid   = threadIdx.x;

#if HAVE_TDM
  // ---- Tensor Data Mover staging: pad_enable reproduces the stride-36 LDS layout in HW ----
  if (tid < 32) {
    // wave 0: 3-D input slab (X = 32 ch, Y = cols, Z = 3 rows), 4B elements,
    // pad 4 dwords after every 32 dwords -> LDS dword stride 36
    unsigned ldsX = (unsigned)(uintptr_t)(void*)sX;
    unsigned long long ga =
        (unsigned long long)(uintptr_t)(inp + (((n * IH + ho) * IW + c0) << 5));
    int tile1 = (wquad == 0) ? TILE_COLS : 64;   // edge strip: stay inside the tensor
    u32x4 g0 = {1u, ldsX, (unsigned)ga, (unsigned)(ga >> 32) | (2u << 30)};
    i32x8 g1;
    g1[0] = (2 << 16) | (1 << 20) | (4 << 22) | (3 << 25); // ds=4B, pad_en, every 32dw, +4dw
    g1[1] = (int)(32u << 16);                    // tensor_dim0 = 32
    g1[2] = (int)(128u << 16);                   // tensor_dim1 = 128
    g1[3] = (int)(32u << 16);                    // tile_dim0  = 32
    g1[4] = (3 << 16) | tile1;                   // tile_dim2 = 3, tile_dim1 = cols
    g1[5] = 32;                                  // tensor_dim0_stride = 32
    g1[6] = (int)(4096u << 16);                  // tensor_dim1_stride = 4096 (low 16 bits)
    g1[7] = 0;
    i32x4 g2 = {128, 0, 0, 0};                   // tensor_dim2 = 128 rows
    i32x4 g3 = {0, 0, 0, 0};
    tdm_load(g0, g1, g2, g3);
    __builtin_amdgcn_s_wait_tensorcnt(0);
  } else if (tid < 64) {
    // wave 1: flat 1-D copy of 4608 x 8B B fragments
    unsigned ldsB = (unsigned)(uintptr_t)(void*)sB;
    unsigned long long gb = (unsigned long long)(uintptr_t)bfrag;
    u32x4 g0 = {1u, ldsB, (unsigned)gb, (unsigned)(gb >> 32) | (2u << 30)};
    i32x8 g1;
    g1[0] = 3 << 16;                             // ds = 8B, no pad
    g1[1] = (int)((4608u & 0xFFFFu) << 16);      // tensor_dim0 = 4608
    g1[2] = (int)((4608u >> 16) | (1u << 16));   // tensor_dim1 = 1
    g1[3] = (int)(4608u << 16);                  // tile_dim0 = 4608
    g1[4] = 1;                                   // tile_dim1 = 1
    g1[5] = 4608;                                // tensor_dim0_stride
    g1[6] = 0;
    g1[7] = 0;
    i32x4 g2 = {1, 0, 0, 0};
    i32x4 g3 = {0, 0, 0, 0};
    tdm_load(g0, g1, g2, g3);
    __builtin_amdgcn_s_wait_tensorcnt(0);
  }
#else
  // ---- fallback staging: explicit global->VGPR->LDS copies ----
  {
    const v4f* src = (const v4f*)bfrag;
    v4f*       dst = (v4f*)sB;
    for (int i = tid; i < KSTEPS * 2 * 32 / 2; i += 128) dst[i] = src[i];
  }
  for (int i = tid; i < 3 * TILE_COLS * 8; i += 128) {
    int ci4 = i & 7;
    int kc  = i >> 3;
    int kh  = kc / TILE_COLS;
    int lc  = kc % TILE_COLS;
    int gc  = c0 + lc; if (gc > IW - 1) gc = IW - 1;
    v4f v = *(const v4f*)(inp + (((n * IH + ho + kh) * IW + gc) << 5) + (ci4 << 2));
    *(v4f*)&sX[(kh * TILE_COLS + lc) * TILE_STRIDE + (ci4 << 2)] = v;
  }
#endif
  __syncthreads();

  int wave  = tid >> 5;
  int lane  = tid & 31;
  int wbl   = wave << 4;          // 0,16,32,48 : wo offset of this wave's 16 M rows
  int m     = lane & 15;          // M row within tile
  int khalf = lane >> 4;          // 0 or 1 -> K sub-pair
  int cioff = khalf << 1;

  v8f acc0 = {};
  v8f acc1 = {};

  #pragma unroll
  for (int ks = 0; ks < KSTEPS; ++ks) {
    const int K0  = ks * 4;
    const int kh  = K0 / 96;
    const int rr  = K0 % 96;
    const int kw  = rr >> 5;
    const int cib = rr & 31;
    int colL = wbl + m + kw;      // column inside the staged slab (<= 65)
    v2f aF = *(const v2f*)&sX[(kh * TILE_COLS + colL) * TILE_STRIDE + cib + cioff];
    v2f b0 = *(const v2f*)&sB[(ks * 2 + 0) * 32 + lane];
    v2f b1 = *(const v2f*)&sB[(ks * 2 + 1) * 32 + lane];
    acc0 = __builtin_amdgcn_wmma_f32_16x16x4_f32(false, aF, false, b0, (short)0, acc0, false, false);
    acc1 = __builtin_amdgcn_wmma_f32_16x16x4_f32(false, aF, false, b1, (short)0, acc1, false, false);
  }

  // gather the 16 y values this lane owns (C layout: VGPR i -> M = i + khalf*8)
  float ys[16], Ry[16], Vp[16], av[16];
  #pragma unroll
  for (int i = 0; i < 8; ++i) { ys[i] = acc0[i]; ys[8 + i] = acc1[i]; }
  #pragma unroll
  for (int i = 0; i < 16; ++i) { Ry[i] = 3000.0f * ys[i]; Vp[i] = 0.0f; av[i] = 0.0f; }

  const float alpha = (float)(0.01 / 15000.0);   // DT/(R*C) with the reference's rounding

  // 200-step LIF recurrence, steps outer / 16 elements inner for ILP.
  // spike update: 14*(sign(v-14)+1) + where(v<14,v,0)  ==  v<14 ? v : (v>14 ? 28 : 14)
  for (int s = 0; s < 200; ++s) {
    #pragma unroll
    for (int i = 0; i < 16; ++i) {
      float v = Vp[i];
      v = (v < 27.0f) ? v : 0.0f;                     // reset after spike
      v = v + (Ry[i] - v) * alpha;                    // Euler step
      v = (v < 14.0f) ? v : ((v > 14.0f) ? 28.0f : 14.0f);
      Vp[i] = v;
      av[i] += v;
    }
  }

  // store: forward value is (Vsum - y) + y, same f32 rounding as the reference
  int baseRow = (n * HO + ho) * WO;
  #pragma unroll
  for (int i = 0; i < 16; ++i) {
    int nt = i >> 3;
    int ri = i & 7;
    int M  = ri + (khalf << 3);
    int wo = c0 + wbl + M;
    if (wo < WO) {
      int N = (lane & 15) + nt * 16;
      out[(baseRow + wo) * CO + N] = (av[i] - ys[i]) + ys[i];
    }
  }
}

extern "C" void kernel_launch(void* const* d_in, const int* in_sizes, int n_in,
                              void* d_out, int out_size, void* d_ws, size_t ws_size,
                              hipStream_t stream) {
  const float* inp = (const float*)d_in[0];   // (16,128,128,32) f32
  const float* kk  = (const float*)d_in[1];   // (3,3,32,32) f32
  float* out = (float*)d_out;                 // (16,126,126,32) f32
  v2f* bfrag = (v2f*)d_ws;                    // 4608 float2 = 36,864 B

  wring_pack_kernel<<<36, 128, 0, stream>>>(kk, bfrag);
  conv_snn_kernel<<<16 * HO * 2, 128, 0, stream>>>(inp, bfrag, out);
}